// Skipgram_55834574848329
// MI455X (gfx1250) — compile-verified
//
#include <hip/hip_runtime.h>
#include <hip/hip_bf16.h>
#include <math.h>

// ---------------------------------------------------------------------------
// Skip-gram negative sampling loss, MI455X (gfx1250, wave32).
//   B=4096 batch, D=128 dim, K=5 negatives.
//   Memory bound: ~14.7 MB of 512B row gathers -> one wave per batch row,
//   one float4 (GLOBAL_LOAD_B128) per lane per row. Cross-lane reduction of
//   the 6 per-lane dot partials done with V_WMMA_F32_16X16X4_F32 (2 scalars
//   reduced per WMMA via a selector B matrix), then LDS atomics, then 6
//   global f32 atomics per block.
// ---------------------------------------------------------------------------

typedef __attribute__((ext_vector_type(2))) float v2f;
typedef __attribute__((ext_vector_type(8))) float v8f;

// Reduce two per-lane f32 values (pa, pb) across all 32 lanes of the wave
// using one V_WMMA_F32_16X16X4_F32, accumulating the two wave-sums into
// ldsacc[qa] and ldsacc[qb].
//
// A (16x4, 2 VGPRs/lane): A[m][0]=pa[lane m], A[m][1]=pb[lane m],
//                         A[m][2]=pa[lane m+16], A[m][3]=pb[lane m+16]
// B (4x16, 2 VGPRs/lane): column 0 = (1,0,1,0)^T  -> selects pa terms
//                         column 1 = (0,1,0,1)^T  -> selects pb terms
// => D[m][0] = pa[m]+pa[m+16], D[m][1] = pb[m]+pb[m+16].
// Summing a lane's 8 D-VGPRs: lane 0 covers rows 0..7 of col 0, lane 16 rows
// 8..15 of col 0 (likewise lanes 1/17 for col 1) -> their sum is the full
// 32-lane reduction.
__device__ __forceinline__ void wmma_reduce_pair(float pa, float pb, int lane,
                                                 float* ldsacc, int qa, int qb) {
    const int nl = lane & 15;
    v2f a; a.x = pa; a.y = pb;
    v2f b; b.x = (nl == 0) ? 1.0f : 0.0f;
    b.y = (nl == 1) ? 1.0f : 0.0f;
    v8f c = {};
    v8f d = __builtin_amdgcn_wmma_f32_16x16x4_f32(
        /*neg_a=*/false, a, /*neg_b=*/false, b,
        /*c_mod=*/(short)0, c, /*reuse_a=*/false, /*reuse_b=*/false);
    float s = d[0] + d[1] + d[2] + d[3] + d[4] + d[5] + d[6] + d[7];
    if (nl == 0) {
        atomicAdd(&ldsacc[qa], s);       // lanes 0 and 16
    } else if (nl == 1) {
        atomicAdd(&ldsacc[qb], s);       // lanes 1 and 17
    }
}

__global__ void sg_init(float* gacc) {
    if (threadIdx.x < 6) gacc[threadIdx.x] = 0.0f;
}

__global__ void __launch_bounds__(256)
sg_main(const int* __restrict__ inw, const int* __restrict__ ctw,
        const int* __restrict__ negw, const float* __restrict__ uw,
        const float* __restrict__ vw, float* __restrict__ gacc, int batch) {
    __shared__ float acc[6];
    const int tid = threadIdx.x;
    if (tid < 6) acc[tid] = 0.0f;
    __syncthreads();

    const int lane = tid & 31;
    const int wave = tid >> 5;
    const int row  = blockIdx.x * 8 + wave;   // one batch row per wave

    // Gather: each lane holds 4 consecutive floats of the 128-float row.
    const int iu = inw[row];
    const int iv = ctw[row];
    float4 u4 = *((const float4*)(uw + (size_t)iu * 128) + lane);
    float4 v4 = *((const float4*)(vw + (size_t)iv * 128) + lane);
    float4 n4[5];
#pragma unroll
    for (int k = 0; k < 5; ++k) {
        const int in = negw[k * batch + row];
        n4[k] = *((const float4*)(vw + (size_t)in * 128) + lane);
    }

    // Per-lane dot partials (4 elements each).
    float pos = u4.x * v4.x + u4.y * v4.y + u4.z * v4.z + u4.w * v4.w;
    float neg[5];
#pragma unroll
    for (int k = 0; k < 5; ++k) {
        neg[k] = u4.x * n4[k].x + u4.y * n4[k].y + u4.z * n4[k].z + u4.w * n4[k].w;
    }

    // Cross-lane reductions: 3 WMMAs reduce all 6 quantities (f32 throughout).
    // All 32 lanes active here (no divergence) -> EXEC all-1s as WMMA requires.
    wmma_reduce_pair(pos,    neg[0], lane, acc, 0, 1);
    wmma_reduce_pair(neg[1], neg[2], lane, acc, 2, 3);
    wmma_reduce_pair(neg[3], neg[4], lane, acc, 4, 5);

    __syncthreads();
    if (tid < 6) atomicAdd(&gacc[tid], acc[tid]);
}

__device__ __forceinline__ float log_sigmoid_f(float x) {
    // Numerically stable: logsig(x) = min(x,0) - log1p(exp(-|x|))
    return fminf(x, 0.0f) - log1pf(expf(-fabsf(x)));
}

__global__ void sg_finish(const float* __restrict__ gacc, float* __restrict__ out) {
    if (threadIdx.x == 0 && blockIdx.x == 0) {
        float lp = log_sigmoid_f(gacc[0]);
        float ln = 0.0f;
#pragma unroll
        for (int k = 0; k < 5; ++k) {
            ln += log_sigmoid_f(-gacc[1 + k]);
        }
        out[0] = -(lp + ln);
    }
}

extern "C" void kernel_launch(void* const* d_in, const int* in_sizes, int n_in,
                              void* d_out, int out_size, void* d_ws, size_t ws_size,
                              hipStream_t stream) {
    const int*   inw  = (const int*)d_in[0];    // input_words   [B] int32
    const int*   ctw  = (const int*)d_in[1];    // context_words [B] int32
    const int*   negw = (const int*)d_in[2];    // neg_words     [K,B] int32
    const float* uw   = (const float*)d_in[3];  // u_weight      [V,128] f32
    const float* vw   = (const float*)d_in[4];  // v_weight      [V,128] f32
    float* gacc = (float*)d_ws;                 // 6 f32 accumulators
    float* out  = (float*)d_out;                // scalar loss

    const int batch = in_sizes[0];              // 4096
    const int blocks = batch / 8;               // 8 waves (rows) per 256-thr block

    sg_init<<<1, 32, 0, stream>>>(gacc);
    sg_main<<<blocks, 256, 0, stream>>>(inw, ctw, negw, uw, vw, gacc, batch);
    sg_finish<<<1, 32, 0, stream>>>(gacc, out);
}